// DSSIML1_33517924778551
// MI455X (gfx1250) — compile-verified
//
#include <hip/hip_runtime.h>
#include <hip/hip_bf16.h>
#include <cstdint>

// Problem shape (fixed by the reference)
#define BATCH 4
#define CHAN  3
#define IMH   1080
#define IMW   1920

// Tile config: 64x24 outputs per 256-thread block (wave32 -> 8 waves)
#define TW 64
#define TH 24
#define LH (TH + 2)            // 26 padded rows
#define LROW 72                // padded LDS row stride in floats (288 B, 16B-aligned)
// LDS row layout: [0..2 pad][3 left-halo][4..67 interior (16B-aligned)][68 right-halo][69..71 pad]
#define COL0 3                 // LDS index of padded column 0
#define PLANE (LROW * LH)      // 1872 floats per plane
#define NPLANES 6              // 3 pred channels + 3 gt channels
#define NROWS (NPLANES * LH)   // 156 LDS rows to fill
#define LDS_WORDS (NPLANES * PLANE)   // 11232 floats = 44928 B

#define SSIM_C1 (0.01f * 0.01f)
#define SSIM_C2 (0.03f * 0.03f)
#define ALPHA_W 0.85f
#define RPT 6                  // output rows per thread (TH / 4)

typedef int v4i __attribute__((vector_size(16)));

typedef __attribute__((address_space(1))) int* gas1_iptr;
typedef __attribute__((address_space(3))) int* as3_iptr;
typedef __attribute__((address_space(1))) v4i* gas1_v4ptr;
typedef __attribute__((address_space(3))) v4i* as3_v4ptr;

__device__ __forceinline__ void async_copy_b32(const float* gsrc, float* lds_dst) {
#if __has_builtin(__builtin_amdgcn_global_load_async_to_lds_b32)
    __builtin_amdgcn_global_load_async_to_lds_b32(
        (gas1_iptr)gsrc, (as3_iptr)lds_dst, /*offset=*/0, /*cpol=*/0);
#else
    unsigned lds_addr = (unsigned)(uintptr_t)lds_dst;
    unsigned long long gaddr = (unsigned long long)(uintptr_t)gsrc;
    asm volatile("global_load_async_to_lds_b32 %0, %1, off"
                 :: "v"(lds_addr), "v"(gaddr) : "memory");
#endif
}

__device__ __forceinline__ void async_copy_b128(const float* gsrc, float* lds_dst) {
#if __has_builtin(__builtin_amdgcn_global_load_async_to_lds_b128)
    __builtin_amdgcn_global_load_async_to_lds_b128(
        (gas1_v4ptr)gsrc, (as3_v4ptr)lds_dst, /*offset=*/0, /*cpol=*/0);
#else
    unsigned lds_addr = (unsigned)(uintptr_t)lds_dst;
    unsigned long long gaddr = (unsigned long long)(uintptr_t)gsrc;
    asm volatile("global_load_async_to_lds_b128 %0, %1, off"
                 :: "v"(lds_addr), "v"(gaddr) : "memory");
#endif
}

__device__ __forceinline__ void wait_async_zero() {
#if __has_builtin(__builtin_amdgcn_s_wait_asynccnt)
    __builtin_amdgcn_s_wait_asynccnt(0);
#else
    asm volatile("s_wait_asynccnt 0" ::: "memory");
#endif
}

__device__ __forceinline__ float fast_rcp(float x) {
#if __has_builtin(__builtin_amdgcn_rcpf)
    return __builtin_amdgcn_rcpf(x);   // v_rcp_f32, ~1 ulp
#else
    return 1.0f / x;
#endif
}

__device__ __forceinline__ int reflect1(int v, int n) {
    // reflect padding of width 1:  -1 -> 1,  n -> n-2
    v = (v < 0) ? -v : v;
    v = (v >= n) ? (2 * n - 2 - v) : v;
    return v;
}

// Horizontal 3-tap sums for one padded row (5 moment sums + center values)
__device__ __forceinline__ void hsum_row(const float* __restrict__ xp,
                                         const float* __restrict__ yp,
                                         int prbase,
                                         float& hx, float& hy,
                                         float& hxx, float& hyy, float& hxy,
                                         float& cx, float& cy) {
    const float x0 = xp[prbase], x1 = xp[prbase + 1], x2 = xp[prbase + 2];
    const float y0 = yp[prbase], y1 = yp[prbase + 1], y2 = yp[prbase + 2];
    hx  = x0 + x1 + x2;
    hy  = y0 + y1 + y2;
    hxx = fmaf(x2, x2, fmaf(x1, x1, x0 * x0));
    hyy = fmaf(y2, y2, fmaf(y1, y1, y0 * y0));
    hxy = fmaf(x2, y2, fmaf(x1, y1, x0 * y0));
    cx  = x1;
    cy  = y1;
}

__global__ __launch_bounds__(256)
void dssim_l1_kernel(const float* __restrict__ pred,
                     const float* __restrict__ gt,
                     float* __restrict__ out) {
    __shared__ float lds[LDS_WORDS];

    const int b  = blockIdx.z;
    const int h0 = blockIdx.y * TH;
    const int w0 = blockIdx.x * TW;
    const int tid = threadIdx.x;

    // ---- Stage 1a: async B128 staging of row interiors (16 chunks of 4 floats per row) ----
    for (int u = tid; u < NROWS * 16; u += 256) {
        const int row = u >> 4;                  // 0..155
        const int k   = u & 15;                  // chunk within row
        const int p   = row / LH;                // plane 0..5
        const int r   = row - p * LH;            // padded row 0..25
        const int gh  = reflect1(h0 + r - 1, IMH);
        const int ch  = (p < CHAN) ? p : (p - CHAN);
        const float* src = (p < CHAN) ? pred : gt;
        const float* grow = src + (((size_t)b * CHAN + ch) * IMH + gh) * (size_t)IMW;
        async_copy_b128(grow + w0 + k * 4,
                        &lds[p * PLANE + r * LROW + (COL0 + 1) + k * 4]);
    }

    // ---- Stage 1b: async B32 staging of the two reflected halo columns per row ----
    for (int u = tid; u < NROWS * 2; u += 256) {
        const int row  = u >> 1;                 // 0..155
        const int side = u & 1;                  // 0 = left, 1 = right
        const int p    = row / LH;
        const int r    = row - p * LH;
        const int gh   = reflect1(h0 + r - 1, IMH);
        const int ch   = (p < CHAN) ? p : (p - CHAN);
        const float* src = (p < CHAN) ? pred : gt;
        const float* grow = src + (((size_t)b * CHAN + ch) * IMH + gh) * (size_t)IMW;
        const int gw   = side ? reflect1(w0 + TW, IMW) : reflect1(w0 - 1, IMW);
        const int lcol = side ? (COL0 + TW + 1) : COL0;
        async_copy_b32(grow + gw, &lds[p * PLANE + r * LROW + lcol]);
    }

    wait_async_zero();
    __syncthreads();

    // ---- Stage 2: separable 3x3 SSIM + L1 with rolling horizontal sums ----
    const int tx   = tid & (TW - 1);         // 0..63
    const int ty   = tid >> 6;               // 0..3
    const int row0 = ty * RPT;               // first output row of this thread (0,6,12,18)
    const float inv9 = 1.0f / 9.0f;
    const float inv3 = 1.0f / 3.0f;

    float acc_ssim[RPT];
    float acc_l1[RPT];
#pragma unroll
    for (int i = 0; i < RPT; ++i) { acc_ssim[i] = 0.0f; acc_l1[i] = 0.0f; }

#pragma unroll
    for (int c = 0; c < CHAN; ++c) {
        const float* xp = &lds[c * PLANE];
        const float* yp = &lds[(c + CHAN) * PLANE];
        const int colbase = tx + COL0;

        // rolling 3-deep horizontal-sum buffers
        float hx[3], hy[3], hxx[3], hyy[3], hxy[3], cx[3], cy[3];
        hsum_row(xp, yp, (row0 + 0) * LROW + colbase,
                 hx[0], hy[0], hxx[0], hyy[0], hxy[0], cx[0], cy[0]);
        hsum_row(xp, yp, (row0 + 1) * LROW + colbase,
                 hx[1], hy[1], hxx[1], hyy[1], hxy[1], cx[1], cy[1]);

#pragma unroll
        for (int i = 0; i < RPT; ++i) {
            const int s2 = (i + 2) % 3;          // slot for new row (compile-time after unroll)
            const int sc = (i + 1) % 3;          // slot holding center row of this window
            hsum_row(xp, yp, (row0 + i + 2) * LROW + colbase,
                     hx[s2], hy[s2], hxx[s2], hyy[s2], hxy[s2], cx[s2], cy[s2]);

            const float sx  = hx[0]  + hx[1]  + hx[2];
            const float sy  = hy[0]  + hy[1]  + hy[2];
            const float sxx = hxx[0] + hxx[1] + hxx[2];
            const float syy = hyy[0] + hyy[1] + hyy[2];
            const float sxy = hxy[0] + hxy[1] + hxy[2];

            const float mu_x = sx * inv9;
            const float mu_y = sy * inv9;
            const float sigma_x  = fmaf(sxx, inv9, -mu_x * mu_x);
            const float sigma_y  = fmaf(syy, inv9, -mu_y * mu_y);
            const float sigma_xy = fmaf(sxy, inv9, -mu_x * mu_y);

            const float n = fmaf(2.0f * mu_x, mu_y, SSIM_C1) * fmaf(2.0f, sigma_xy, SSIM_C2);
            const float d = (fmaf(mu_x, mu_x, mu_y * mu_y) + SSIM_C1) *
                            (sigma_x + sigma_y + SSIM_C2);
            float s = fmaf(-0.5f * n, fast_rcp(d), 0.5f);   // (1 - n/d) * 0.5
            s = fminf(fmaxf(s, 0.0f), 1.0f);
            acc_ssim[i] += s;
            acc_l1[i]   += fabsf(cx[sc] - cy[sc]);
        }
    }

#pragma unroll
    for (int i = 0; i < RPT; ++i) {
        const float val = ALPHA_W * (acc_ssim[i] * inv3) +
                          (1.0f - ALPHA_W) * (acc_l1[i] * inv3);
        out[((size_t)b * IMH + (h0 + row0 + i)) * (size_t)IMW + (w0 + tx)] = val;
    }
}

extern "C" void kernel_launch(void* const* d_in, const int* in_sizes, int n_in,
                              void* d_out, int out_size, void* d_ws, size_t ws_size,
                              hipStream_t stream) {
    (void)in_sizes; (void)n_in; (void)out_size; (void)d_ws; (void)ws_size;
    const float* pred = (const float*)d_in[0];
    const float* gt   = (const float*)d_in[1];
    float* out        = (float*)d_out;

    dim3 grid(IMW / TW, IMH / TH, BATCH);   // 30 x 45 x 4 = 5400 blocks
    dim3 block(256, 1, 1);                  // 8 wave32 waves
    dssim_l1_kernel<<<grid, block, 0, stream>>>(pred, gt, out);
}